// SinusoidalSynthesizer_55301998903903
// MI455X (gfx1250) — compile-verified
//
#include <hip/hip_runtime.h>
#include <math.h>

// ---------------------------------------------------------------------------
// DDSP sinusoidal synthesizer for MI455X (gfx1250, wave32).
//
// Structure:
//   K1 controls: per-element sigmoids -> amp, freq (+nyquist mask), to ws.
//   K2 scan:     per-channel chunk sums (analytic, chunk=256 samples) and the
//                250-chunk exclusive prefix computed with V_WMMA_F32_16X16X4_F32
//                (scan-as-triangular-matmul, 4 chained K=4 WMMAs per channel).
//   K3 synth:    closed-form piecewise-quadratic in-chunk phase (no serial
//                cumsum at all), interpolated amp, sum over 64 oscillators.
// ---------------------------------------------------------------------------

typedef __attribute__((ext_vector_type(2))) float v2f;
typedef __attribute__((ext_vector_type(8))) float v8f;

#define N_B      8
#define N_T      250
#define N_K      64
#define N_CH     (N_B * N_K)     // 512 channels
#define N_SAMP   64000
#define CHUNK    256             // samples per control-knot interval (64000/250)
#define N_CHUNK  250
#define W2PI     3.92699081698724e-4f   // 2*pi/16000
#define LOG10E_  2.302585092994046f     // ln(10)  (exp_sigmoid exponent)
#define MIDI_MAX 119.21315f             // 12*log2(8000/440)+69
#define NYQ      8000.0f

// ---- Kernel 1: get_controls ------------------------------------------------
__global__ void controls_kernel(const float* __restrict__ amp_in,
                                const float* __restrict__ frq_in,
                                float* __restrict__ amps_c,
                                float* __restrict__ freqs_c) {
  int i = blockIdx.x * blockDim.x + threadIdx.x;
  if (i >= N_B * N_T * N_K) return;
  float a = amp_in[i];
  float f = frq_in[i];
  float sa  = 1.0f / (1.0f + expf(-a));
  float amp = 2.0f * powf(sa, LOG10E_) + 1e-7f;          // exp_sigmoid
  float uf  = 1.0f / (1.0f + expf(-f));
  float frq = 440.0f * exp2f((MIDI_MAX * uf - 69.0f) * (1.0f / 12.0f));
  amp = (frq >= NYQ) ? 0.0f : amp;                       // remove_above_nyquist
  amps_c[i]  = amp;
  freqs_c[i] = frq;
}

// ---- Kernel 2: per-channel chunk-prefix via WMMA scan ----------------------
// One wave (32 threads) per channel.  Chunk sum of omega over 256 samples:
//   S_c = (32*f[c-1] + 192*f[c] + 32*f[c+1]) * 2pi/16000   (indices clamped)
// Exclusive prefix over 256 (padded) chunk sums, laid out as 16x16 matrix M:
//   P[i*16+j] = sum_{i'<i} rowtotal(i') + (M @ U_strict)[i][j]
// The row scan is 4 chained v_wmma_f32_16x16x4_f32 ops.
__global__ void scan_kernel(const float* __restrict__ freqs_c,
                            float* __restrict__ P) {
  __shared__ float fsh[256];
  __shared__ float ssh[256];
  __shared__ float Tsh[16];
  int ch   = blockIdx.x;            // 0..511
  int b    = ch >> 6;
  int k    = ch & 63;
  int lane = threadIdx.x;           // 0..31

  // Stage the 250 control frequencies for this channel.
#pragma unroll
  for (int j = 0; j < 8; ++j) {
    int t = lane + 32 * j;
    fsh[t] = (t < N_T) ? freqs_c[(b * N_T + t) * N_K + k] : 0.0f;
  }
  __syncthreads();

  // Analytic chunk sums (exact for half-pixel linear upsample, scale 1/256).
#pragma unroll
  for (int j = 0; j < 8; ++j) {
    int c = lane + 32 * j;
    float s = 0.0f;
    if (c < N_CHUNK) {
      int cm = (c == 0) ? 0 : c - 1;
      int cp = (c == N_CHUNK - 1) ? (N_CHUNK - 1) : c + 1;
      s = (32.0f * fsh[cm] + 192.0f * fsh[c] + 32.0f * fsh[cp]) * W2PI;
    }
    ssh[c] = s;
  }
  __syncthreads();

  // Row totals of the 16x16 matrix (for the cross-row carry).
  if (lane < 16) {
    float t = 0.0f;
#pragma unroll
    for (int j = 0; j < 16; ++j) t += ssh[lane * 16 + j];
    Tsh[lane] = t;
  }
  __syncthreads();

  // Row-wise exclusive scan E = M @ U_strict via 4 chained 16x16x4 f32 WMMAs.
  // A layout (16x4 f32): lanes 0-15 row=lane hold K={0,1}; lanes 16-31 hold K={2,3}.
  // B layout (4x16 f32): lanes 0-15 col=lane hold K={0,1}; lanes 16-31 hold K={2,3}.
  int row = lane & 15;
  int hi  = lane >> 4;              // 0 or 1
  int col = lane & 15;
  v8f acc = {};
#pragma unroll
  for (int kb = 0; kb < 4; ++kb) {
    int k0 = 4 * kb + 2 * hi;
    v2f afrag, bfrag;
    afrag.x = ssh[row * 16 + k0];
    afrag.y = ssh[row * 16 + k0 + 1];
    bfrag.x = (k0     < col) ? 1.0f : 0.0f;   // U_strict[k][j] = (k < j)
    bfrag.y = (k0 + 1 < col) ? 1.0f : 0.0f;
    acc = __builtin_amdgcn_wmma_f32_16x16x4_f32(
        /*neg_a=*/false, afrag, /*neg_b=*/false, bfrag,
        /*c_mod=*/(short)0, acc, /*reuse_a=*/false, /*reuse_b=*/false);
  }

  // C/D layout: acc[v] holds E[v + 8*hi][col].  Add exclusive row-total carry.
#pragma unroll
  for (int v = 0; v < 8; ++v) {
    int i = v + 8 * hi;
    float g = 0.0f;
    for (int ip = 0; ip < i; ++ip) g += Tsh[ip];
    P[ch * CHUNK + i * 16 + col] = acc[v] + g;
  }
}

// ---- Kernel 3: oscillator bank, closed-form phase --------------------------
// One block per (batch, chunk); one thread per output sample (r = tid).
__global__ void synth_kernel(const float* __restrict__ amps_c,
                             const float* __restrict__ freqs_c,
                             const float* __restrict__ P,
                             float* __restrict__ out) {
  __shared__ float fm1[64], f0s[64], fp1[64];
  __shared__ float am1[64], a0s[64], ap1[64];
  __shared__ float Pc[64];
  int bc  = blockIdx.x;
  int b   = bc / N_CHUNK;
  int c   = bc % N_CHUNK;
  int tid = threadIdx.x;            // 0..255

  if (tid < 64) {
    int k  = tid;
    int cm = (c == 0) ? 0 : c - 1;
    int cp = (c == N_CHUNK - 1) ? (N_CHUNK - 1) : c + 1;
    fm1[k] = freqs_c[(b * N_T + cm) * N_K + k];
    f0s[k] = freqs_c[(b * N_T + c ) * N_K + k];
    fp1[k] = freqs_c[(b * N_T + cp) * N_K + k];
    am1[k] = amps_c [(b * N_T + cm) * N_K + k];
    a0s[k] = amps_c [(b * N_T + c ) * N_K + k];
    ap1[k] = amps_c [(b * N_T + cp) * N_K + k];
    Pc[k]  = P[(b * N_K + k) * CHUNK + c];
  }
  __syncthreads();

  float r     = (float)tid;
  bool  first = tid < 128;          // wave-uniform on wave32
  float acc   = 0.0f;

#pragma unroll 4
  for (int k = 0; k < 64; ++k) {
    float fa = fm1[k], fb = f0s[k], fc = fp1[k];
    float cum, w, fl, fh, al, ah;
    if (first) {
      // x = c-1 + (r+128.5)/256 ; interpolate knots (c-1, c)
      w   = (r + 128.5f) * (1.0f / 256.0f);
      cum = (r + 1.0f) * fa +
            (fb - fa) * ((r + 1.0f) * (r + 257.0f) * (1.0f / 512.0f));
      fl = fa; fh = fb; al = am1[k]; ah = a0s[k];
    } else {
      // x = c + (m+0.5)/256, m = r-128 ; interpolate knots (c, c+1)
      float m = r - 128.0f;
      w   = (m + 0.5f) * (1.0f / 256.0f);
      cum = 32.0f * fa + 96.0f * fb + (m + 1.0f) * fb +
            (fc - fb) * ((m + 1.0f) * (m + 1.0f) * (1.0f / 512.0f));
      fl = fb; fh = fc; al = a0s[k]; ah = ap1[k];
    }
    float amp = al + w * (ah - al);
    float fe  = fl + w * (fh - fl);
    amp = (fe >= NYQ) ? 0.0f : amp;            // nyquist mask on resampled freq
    float phase = Pc[k] + cum * W2PI;
    acc += amp * sinf(phase);
  }
  out[b * N_SAMP + c * CHUNK + tid] = acc;
}

// ---------------------------------------------------------------------------
extern "C" void kernel_launch(void* const* d_in, const int* in_sizes, int n_in,
                              void* d_out, int out_size, void* d_ws, size_t ws_size,
                              hipStream_t stream) {
  const float* amp_in = (const float*)d_in[0];
  const float* frq_in = (const float*)d_in[1];
  float* out = (float*)d_out;

  float* wsf     = (float*)d_ws;
  float* freqs_c = wsf;                          // 128000 f32
  float* amps_c  = wsf + N_B * N_T * N_K;        // 128000 f32
  float* P       = wsf + 2 * N_B * N_T * N_K;    // 512*256  f32  (~1.5 MB total)

  controls_kernel<<<(N_B * N_T * N_K + 255) / 256, 256, 0, stream>>>(
      amp_in, frq_in, amps_c, freqs_c);
  scan_kernel<<<N_CH, 32, 0, stream>>>(freqs_c, P);
  synth_kernel<<<N_B * N_CHUNK, 256, 0, stream>>>(amps_c, freqs_c, P, out);
}